// HeteroGraphSAGE_23570780520593
// MI455X (gfx1250) — compile-verified
//
#include <hip/hip_runtime.h>

#define N_USER 50000
#define N_TX   100000
#define NEDGE  600000
#define D_IN   128

typedef __attribute__((ext_vector_type(2))) float v2f;
typedef __attribute__((ext_vector_type(8))) float v8f;

// ---------------------------------------------------------------------------
// Native CDNA5 fp32 global atomic add (no CAS loop).
// ---------------------------------------------------------------------------
__device__ __forceinline__ void gatomic_fadd(float* p, float v) {
    asm volatile("global_atomic_add_f32 %0, %1, off" : : "v"(p), "v"(v) : "memory");
}

// ---------------------------------------------------------------------------
// Edge scatter: agg[dst] += src_feat[src], 4 floats per thread (float4 load).
// E*32 threads.
// ---------------------------------------------------------------------------
__global__ void scatter_feat_kernel(const float* __restrict__ src_feat,
                                    const int* __restrict__ src_idx,
                                    const int* __restrict__ dst_idx,
                                    float* __restrict__ agg) {
    int tid = blockIdx.x * blockDim.x + threadIdx.x;
    if (tid >= NEDGE * 32) return;
    int e = tid >> 5;
    int c = (tid & 31) << 2;
    int s = src_idx[e];
    int d = dst_idx[e];
    float4 v = *(const float4*)(src_feat + (size_t)s * D_IN + c);
    float* p = agg + (size_t)d * D_IN + c;
    gatomic_fadd(p + 0, v.x);
    gatomic_fadd(p + 1, v.y);
    gatomic_fadd(p + 2, v.z);
    gatomic_fadd(p + 3, v.w);
}

__global__ void scatter_deg_kernel(const int* __restrict__ dst_idx,
                                   float* __restrict__ deg) {
    int e = blockIdx.x * blockDim.x + threadIdx.x;
    if (e >= NEDGE) return;
    gatomic_fadd(&deg[dst_idx[e]], 1.0f);
}

// agg[i][:] /= max(deg[i], 1)
__global__ void mean_div_kernel(float* __restrict__ agg,
                                const float* __restrict__ deg,
                                int ndst) {
    int tid = blockIdx.x * blockDim.x + threadIdx.x;
    if (tid >= ndst * 32) return;
    int i = tid >> 5;
    int c = (tid & 31) << 2;
    float d = fmaxf(deg[i], 1.0f);
    float4 v = *(float4*)(agg + (size_t)i * D_IN + c);
    v.x = v.x / d; v.y = v.y / d; v.z = v.z / d; v.w = v.w / d;
    *(float4*)(agg + (size_t)i * D_IN + c) = v;
}

// ---------------------------------------------------------------------------
// Fused SAGE GEMM: Out = relu?( scale * (Hdst @ Wself^T + Agg @ Wneigh^T + b)
//                               [+ Out if accum] )
// Full fp32 via V_WMMA_F32_16X16X4_F32. One 16-row band per block; one wave
// per 16-col tile. A-tiles (Hdst, Agg) staged in LDS; W rows stream from L2.
//
// 32-bit A 16x4 layout : lane m=lane&15, K pair at kbase + 2*(lane>>4)
// 32-bit B 4x16 layout : lane n=lane&15, K pair at kbase + 2*(lane>>4)
// 32-bit C/D 16x16     : VGPR r -> row r + 8*(lane>>4), col lane&15
// ---------------------------------------------------------------------------
template <int DOUT>
__global__ __launch_bounds__(32 * (DOUT / 16))
void sage_wmma_kernel(const float* __restrict__ Hdst,
                      const float* __restrict__ Agg,
                      const float* __restrict__ Wself,
                      const float* __restrict__ Wneigh,
                      const float* __restrict__ bias,
                      float* __restrict__ Out,
                      float scale, int accum, int relu) {
    __shared__ float sH[16][D_IN + 4];
    __shared__ float sA[16][D_IN + 4];

    const int row0 = blockIdx.x * 16;

    // cooperative stage of the two 16x128 A-tiles
    for (int idx = threadIdx.x; idx < 16 * (D_IN / 4); idx += blockDim.x) {
        int r = idx >> 5;            // D_IN/4 == 32 chunks per row
        int c = (idx & 31) << 2;
        *(float4*)&sH[r][c] = *(const float4*)(Hdst + (size_t)(row0 + r) * D_IN + c);
        *(float4*)&sA[r][c] = *(const float4*)(Agg  + (size_t)(row0 + r) * D_IN + c);
    }
    __syncthreads();

    const int lane = threadIdx.x & 31;
    const int wave = threadIdx.x >> 5;
    const int colb = wave * 16;
    const int m    = lane & 15;       // A row within tile / B col within tile
    const int g    = lane >> 4;       // K half-select
    const int n    = colb + m;        // output column

    const float* wsrow = Wself  + (size_t)n * D_IN;   // B[k][n] = W[n][k]
    const float* wnrow = Wneigh + (size_t)n * D_IN;
    const float  bn    = bias[n];

    v8f acc = {0.f, 0.f, 0.f, 0.f, 0.f, 0.f, 0.f, 0.f};

#pragma unroll
    for (int kk = 0; kk < D_IN; kk += 4) {
        const int k = kk + 2 * g;
        v2f a  = *(const v2f*)&sH[m][k];
        v2f b  = *(const v2f*)&wsrow[k];
        acc = __builtin_amdgcn_wmma_f32_16x16x4_f32(false, a, false, b,
                                                    (short)0, acc, false, false);
        v2f a2 = *(const v2f*)&sA[m][k];
        v2f b2 = *(const v2f*)&wnrow[k];
        acc = __builtin_amdgcn_wmma_f32_16x16x4_f32(false, a2, false, b2,
                                                    (short)0, acc, false, false);
    }

#pragma unroll
    for (int r = 0; r < 8; ++r) {
        int mo = row0 + r + 8 * g;
        float v = (acc[r] + bn) * scale;
        float* o = Out + (size_t)mo * DOUT + n;
        if (accum) v += *o;
        if (relu)  v = fmaxf(v, 0.f);
        *o = v;
    }
}

// ---------------------------------------------------------------------------
// Host orchestration (all on `stream`, graph-capture safe).
// ---------------------------------------------------------------------------
static void run_mean_agg(const float* feat, const int* src, const int* dst,
                         float* agg, float* deg, int ndst, hipStream_t stream) {
    hipMemsetAsync(agg, 0, (size_t)ndst * D_IN * sizeof(float), stream);
    hipMemsetAsync(deg, 0, (size_t)ndst * sizeof(float), stream);
    scatter_feat_kernel<<<(NEDGE * 32 + 255) / 256, 256, 0, stream>>>(feat, src, dst, agg);
    scatter_deg_kernel<<<(NEDGE + 255) / 256, 256, 0, stream>>>(dst, deg);
    mean_div_kernel<<<(ndst * 32 + 255) / 256, 256, 0, stream>>>(agg, deg, ndst);
}

extern "C" void kernel_launch(void* const* d_in, const int* in_sizes, int n_in,
                              void* d_out, int out_size, void* d_ws, size_t ws_size,
                              hipStream_t stream) {
    const float* x_user      = (const float*)d_in[0];
    const float* x_tx        = (const float*)d_in[1];
    const int*   makes_src   = (const int*)d_in[2];
    const int*   makes_dst   = (const int*)d_in[3];
    const int*   made_by_src = (const int*)d_in[4];
    const int*   made_by_dst = (const int*)d_in[5];
    const int*   refers_src  = (const int*)d_in[6];
    const int*   refers_dst  = (const int*)d_in[7];

    // params flattened in dict order: for l in {0,1}: for e in {makes, made_by,
    // refers}: Wself, Wneigh, b
    const float* P[18];
    for (int i = 0; i < 18; ++i) P[i] = (const float*)d_in[8 + i];
    // l0: makes P[0..2], made_by P[3..5], refers P[6..8]
    // l1: makes P[9..11], made_by P[12..14], refers P[15..17]

    // workspace carve-up
    float* agg_tx = (float*)d_ws;                       // N_TX  * 128
    float* agg_u  = agg_tx + (size_t)N_TX * D_IN;       // N_USER* 128 (reused)
    float* deg_tx = agg_u + (size_t)N_USER * D_IN;      // N_TX
    float* deg_u  = deg_tx + N_TX;                      // N_USER
    float* h_tx   = deg_u + N_USER;                     // N_TX  * 128
    float* h_u    = h_tx + (size_t)N_TX * D_IN;         // N_USER* 128

    float* out_user = (float*)d_out;                    // [N_USER, 64]
    float* out_tx   = out_user + (size_t)N_USER * 64;   // [N_TX, 64]

    // ---------------- Layer 0 (din=128, dout=128), ReLU after etype-mean ----
    // makes: user -> tx
    run_mean_agg(x_user, makes_src, makes_dst, agg_tx, deg_tx, N_TX, stream);
    sage_wmma_kernel<128><<<N_TX / 16, 256, 0, stream>>>(
        x_tx, agg_tx, P[0], P[1], P[2], h_tx, 1.0f, 0, 1);

    // made_by: tx -> user (first half of the 0.5*(u1+u2) mean)
    run_mean_agg(x_tx, made_by_src, made_by_dst, agg_u, deg_u, N_USER, stream);
    sage_wmma_kernel<128><<<N_USER / 16, 256, 0, stream>>>(
        x_user, agg_u, P[3], P[4], P[5], h_u, 0.5f, 0, 0);

    // refers: user -> user (accumulate second half, then ReLU)
    run_mean_agg(x_user, refers_src, refers_dst, agg_u, deg_u, N_USER, stream);
    sage_wmma_kernel<128><<<N_USER / 16, 256, 0, stream>>>(
        x_user, agg_u, P[6], P[7], P[8], h_u, 0.5f, 1, 1);

    // ---------------- Layer 1 (din=128, dout=64), no activation -------------
    run_mean_agg(h_u, makes_src, makes_dst, agg_tx, deg_tx, N_TX, stream);
    sage_wmma_kernel<64><<<N_TX / 16, 128, 0, stream>>>(
        h_tx, agg_tx, P[9], P[10], P[11], out_tx, 1.0f, 0, 0);

    run_mean_agg(h_tx, made_by_src, made_by_dst, agg_u, deg_u, N_USER, stream);
    sage_wmma_kernel<64><<<N_USER / 16, 128, 0, stream>>>(
        h_u, agg_u, P[12], P[13], P[14], out_user, 0.5f, 0, 0);

    run_mean_agg(h_u, refers_src, refers_dst, agg_u, deg_u, N_USER, stream);
    sage_wmma_kernel<64><<<N_USER / 16, 128, 0, stream>>>(
        h_u, agg_u, P[15], P[16], P[17], out_user, 0.5f, 1, 0);
}